// FastAttention_43808666419232
// MI455X (gfx1250) — compile-verified
//
#include <hip/hip_runtime.h>

// MI455X / gfx1250, wave32. K1/K3 matmuls via V_WMMA_F32_16X16X4_F32; K2 uses scalar loops.

typedef __attribute__((ext_vector_type(2))) float v2f;
typedef __attribute__((ext_vector_type(8))) float v8f;

#define BH 64          // B*H
#define NN 4096        // sequence
#define DD 64          // head dim (= nb_features after Gram collapse)
#define K1_ROWS 256
#define K1_CHUNKS 16   // NN / K1_ROWS
#define C64 64         // NN / 64 : scan granularity & K3 chunks

static __device__ __forceinline__ v8f wmma_f32(v2f a, v2f b, v8f c) {
  // 8 args: (neg_a, A, neg_b, B, c_mod, C, reuse_a, reuse_b)
  return __builtin_amdgcn_wmma_f32_16x16x4_f32(false, a, false, b, (short)0, c,
                                               false, false);
}

// ---------------------------------------------------------------------------
// K1: per (bh, 256-row chunk):
//   Gpart[bh][chunk] = k_chunk^T @ v_chunk   (64x64, WMMA f32)
//   S[bh][c64][d]    = 64-row column sums of k (for cross-chunk cumsum)
// ---------------------------------------------------------------------------
__global__ void k1_gpart(const float* __restrict__ k, const float* __restrict__ v,
                         float* __restrict__ Gpart, float* __restrict__ S) {
  const int bh = blockIdx.x;
  const int ch = blockIdx.y;
  const int t  = threadIdx.x;
  const float* kb = k + (size_t)bh * NN * DD;
  const float* vb = v + (size_t)bh * NN * DD;
  const int n0 = ch * K1_ROWS;

  // 64-row column sums of k
  {
    const int d = t & 63, seg = t >> 6;
    const int r0 = n0 + seg * 64;
    float s = 0.f;
    for (int r = 0; r < 64; ++r) s += kb[(size_t)(r0 + r) * DD + d];
    S[((size_t)(bh * C64) + (ch * 4 + seg)) * DD + d] = s;
  }

  // G partial via WMMA: A = k^T (DxK over rows), B = v
  const int wave = t >> 5, lane = t & 31;
  const int m = lane & 15, khalf = (lane >> 4) * 2;
  float* gout = Gpart + (size_t)(bh * K1_CHUNKS + ch) * (DD * DD);
  for (int i = 0; i < 2; ++i) {
    const int tt = wave * 2 + i;
    const int ti = tt >> 2, tj = tt & 3;      // d-tile, e-tile
    v8f acc = {0.f, 0.f, 0.f, 0.f, 0.f, 0.f, 0.f, 0.f};
    for (int kk = 0; kk < K1_ROWS; kk += 4) {
      const int ra = n0 + kk + khalf;
      v2f a, b;
      a.x = kb[(size_t)ra * DD + ti * 16 + m];        // A[m][khalf]   = k[ra][d0+m]
      a.y = kb[(size_t)(ra + 1) * DD + ti * 16 + m];  // A[m][khalf+1]
      b.x = vb[(size_t)ra * DD + tj * 16 + m];        // B[khalf][m]   = v[ra][e0+m]
      b.y = vb[(size_t)(ra + 1) * DD + tj * 16 + m];
      acc = wmma_f32(a, b, acc);
    }
    const int rbase = ti * 16 + (lane >> 4) * 8;
    for (int j = 0; j < 8; ++j)
      gout[(rbase + j) * DD + tj * 16 + m] = acc[j];
  }
}

// ---------------------------------------------------------------------------
// K2: per bh: G = sum(Gpart); Gram = P^T P; W = Gram @ G; exclusive scan of S.
// ---------------------------------------------------------------------------
__global__ void k2_combine(const float* __restrict__ proj,
                           const float* __restrict__ Gpart,
                           float* __restrict__ S, float* __restrict__ Wm,
                           float* __restrict__ GramG) {
  __shared__ float G[DD * DD];
  __shared__ float Pl[DD * DD];
  __shared__ float Gm[DD * DD];
  const int bh = blockIdx.x;
  const int t  = threadIdx.x;

  for (int idx = t; idx < DD * DD; idx += 256) {
    float s = 0.f;
    for (int c = 0; c < K1_CHUNKS; ++c)
      s += Gpart[(size_t)(bh * K1_CHUNKS + c) * (DD * DD) + idx];
    G[idx]  = s;
    Pl[idx] = proj[idx];
  }
  __syncthreads();

  for (int idx = t; idx < DD * DD; idx += 256) {
    const int d1 = idx >> 6, d2 = idx & 63;
    float s = 0.f;
    for (int mm = 0; mm < DD; ++mm) s += Pl[mm * DD + d1] * Pl[mm * DD + d2];
    Gm[idx] = s;
    if (bh == 0) GramG[idx] = s;  // one block publishes Gram for K3
  }
  __syncthreads();

  for (int idx = t; idx < DD * DD; idx += 256) {
    const int d1 = idx >> 6, e = idx & 63;
    float s = 0.f;
    for (int d2 = 0; d2 < DD; ++d2) s += Gm[d1 * DD + d2] * G[d2 * DD + e];
    Wm[(size_t)bh * DD * DD + idx] = s;
  }

  // exclusive scan of 64-row sums over the sequence (in place)
  if (t < 64) {
    float run = 0.f;
    for (int c = 0; c < C64; ++c) {
      const size_t idx = ((size_t)(bh * C64) + c) * DD + t;
      const float tmp = S[idx];
      S[idx] = run;
      run += tmp;
    }
  }
}

// ---------------------------------------------------------------------------
// K3: per (bh, 64-row chunk):
//   c = prefix + local inclusive cumsum of k rows (D-space)
//   qg = q @ Gram (WMMA);  den[r] = dot(qg[r], c[r]);  out = (q @ W) / den
// ---------------------------------------------------------------------------
__global__ void k3_out(const float* __restrict__ q, const float* __restrict__ k,
                       const float* __restrict__ S, const float* __restrict__ Wm,
                       const float* __restrict__ GramG, float* __restrict__ out) {
  __shared__ float kc[64 * 65];    // k cumsum (padded vs 64-bank LDS)
  __shared__ float ab[64 * 65];    // Gram, later reused for W
  __shared__ float qgl[64 * 65];   // qg = q @ Gram
  __shared__ float qsum[4 * 64];
  __shared__ float dinv[64];
  const int bh = blockIdx.x, c64 = blockIdx.y;
  const int t  = threadIdx.x;
  const int n0 = c64 * 64;
  const float* qb = q + (size_t)bh * NN * DD;
  const float* kb = k + (size_t)bh * NN * DD;

  for (int idx = t; idx < 64 * 64; idx += 256) {
    const int r = idx >> 6, d = idx & 63;
    kc[r * 65 + d] = kb[(size_t)n0 * DD + idx];
    ab[r * 65 + d] = GramG[idx];
  }
  __syncthreads();

  // two-level inclusive cumsum over 64 rows, per column d
  const int d = t & 63, qt = t >> 6;
  {
    float run = 0.f;
    for (int r = qt * 16; r < qt * 16 + 16; ++r) {
      run += kc[r * 65 + d];
      kc[r * 65 + d] = run;
    }
    qsum[qt * 64 + d] = run;
  }
  __syncthreads();
  if (t < 64) {
    const float p = S[((size_t)(bh * C64) + c64) * DD + t];  // exclusive prefix
    const float o0 = p;
    const float o1 = o0 + qsum[0 * 64 + t];
    const float o2 = o1 + qsum[1 * 64 + t];
    const float o3 = o2 + qsum[2 * 64 + t];
    qsum[0 * 64 + t] = o0; qsum[1 * 64 + t] = o1;
    qsum[2 * 64 + t] = o2; qsum[3 * 64 + t] = o3;
  }
  __syncthreads();
  {
    const float off = qsum[qt * 64 + d];
    for (int r = qt * 16; r < qt * 16 + 16; ++r) kc[r * 65 + d] += off;
  }
  __syncthreads();

  const int wave = t >> 5, lane = t & 31;
  const int m = lane & 15, khalf = (lane >> 4) * 2;

  // qg = q_chunk @ Gram
  for (int i = 0; i < 2; ++i) {
    const int tt = wave * 2 + i;
    const int ti = tt >> 2, tj = tt & 3;
    v8f acc = {0.f, 0.f, 0.f, 0.f, 0.f, 0.f, 0.f, 0.f};
    const float* arow = qb + (size_t)(n0 + ti * 16 + m) * DD;
    for (int kk = 0; kk < DD; kk += 4) {
      v2f a = *(const v2f*)(arow + kk + khalf);         // 8B-aligned float2
      v2f b;
      b.x = ab[(kk + khalf) * 65 + tj * 16 + m];
      b.y = ab[(kk + khalf + 1) * 65 + tj * 16 + m];
      acc = wmma_f32(a, b, acc);
    }
    const int rbase = ti * 16 + (lane >> 4) * 8;
    for (int j = 0; j < 8; ++j)
      qgl[(rbase + j) * 65 + tj * 16 + m] = acc[j];
  }
  __syncthreads();

  if (t < 64) {
    float s = 0.f;
    for (int dd = 0; dd < 64; ++dd) s += qgl[t * 65 + dd] * kc[t * 65 + dd];
    dinv[t] = 1.0f / s;
  }
  __syncthreads();

  // reuse ab for W
  for (int idx = t; idx < 64 * 64; idx += 256) {
    const int r = idx >> 6, dd = idx & 63;
    ab[r * 65 + dd] = Wm[(size_t)bh * DD * DD + idx];
  }
  __syncthreads();

  // out = (q_chunk @ W) * dinv
  for (int i = 0; i < 2; ++i) {
    const int tt = wave * 2 + i;
    const int ti = tt >> 2, tj = tt & 3;
    v8f acc = {0.f, 0.f, 0.f, 0.f, 0.f, 0.f, 0.f, 0.f};
    const float* arow = qb + (size_t)(n0 + ti * 16 + m) * DD;
    for (int kk = 0; kk < DD; kk += 4) {
      v2f a = *(const v2f*)(arow + kk + khalf);
      v2f b;
      b.x = ab[(kk + khalf) * 65 + tj * 16 + m];
      b.y = ab[(kk + khalf + 1) * 65 + tj * 16 + m];
      acc = wmma_f32(a, b, acc);
    }
    const int rbase = ti * 16 + (lane >> 4) * 8;
    for (int j = 0; j < 8; ++j) {
      const float dv = dinv[rbase + j];
      out[((size_t)bh * NN + n0 + rbase + j) * DD + tj * 16 + m] = acc[j] * dv;
    }
  }
}

// ---------------------------------------------------------------------------
extern "C" void kernel_launch(void* const* d_in, const int* in_sizes, int n_in,
                              void* d_out, int out_size, void* d_ws, size_t ws_size,
                              hipStream_t stream) {
  const float* q    = (const float*)d_in[0];
  const float* k    = (const float*)d_in[1];
  const float* v    = (const float*)d_in[2];
  const float* proj = (const float*)d_in[3];
  float* out = (float*)d_out;

  float* ws    = (float*)d_ws;
  float* Gpart = ws;                                        // 64*16*4096
  float* S     = Gpart + (size_t)BH * K1_CHUNKS * DD * DD;  // 64*64*64
  float* Wm    = S + (size_t)BH * C64 * DD;                 // 64*4096
  float* Gram  = Wm + (size_t)BH * DD * DD;                 // 4096

  k1_gpart<<<dim3(BH, K1_CHUNKS), 256, 0, stream>>>(k, v, Gpart, S);
  k2_combine<<<dim3(BH), 256, 0, stream>>>(proj, Gpart, S, Wm, Gram);
  k3_out<<<dim3(BH, C64), 256, 0, stream>>>(q, k, S, Wm, Gram, out);
}